// WeightedBCEWithLogitsLoss_66039417143805
// MI455X (gfx1250) — compile-verified
//
#include <hip/hip_runtime.h>
#include <hip/hip_bf16.h>

// ---------------------------------------------------------------------------
// WeightedBCEWithLogitsLoss for MI455X (gfx1250, wave32)
//   B=32, H=512, W=512, N = 8,388,608 fp32 elements per tensor.
//   Memory-bound: single-pass atomic union-find CCL + fused weighted-BCE
//   reduction. Final cross-block reduction uses v_wmma_f32_16x16x4_f32.
// ---------------------------------------------------------------------------

typedef __attribute__((ext_vector_type(2))) float v2f;
typedef __attribute__((ext_vector_type(8))) float v8f;

#define CCL_W   512
#define CCL_HW  (512 * 512)
#define NB_PART 2048   // partial-sum blocks for k_loss (multiple of 16)

// ---- union-find helpers ----------------------------------------------------
__device__ __forceinline__ int find_root(volatile int* L, int x) {
  int p = L[x];
  while (p != x) { x = p; p = L[x]; }
  return x;
}

__device__ __forceinline__ void unite(int* L, int x, int y) {
  int a = find_root(L, x);
  int b = find_root(L, y);
  while (a != b) {
    int mn = a < b ? a : b;
    int mx = a < b ? b : a;
    int old = atomicMin(&L[mx], mn);
    if (old == mx) break;               // successfully linked mx -> mn
    a = find_root(L, mn);               // contention: retry from new roots
    b = find_root(L, old);
  }
}

// ---- K1: init labels (fg -> own index, bg -> -1) and zero counts -----------
__global__ void k_init(const float* __restrict__ t, int* __restrict__ lab,
                       unsigned* __restrict__ cnt, int nchunks) {
  int c = blockIdx.x * blockDim.x + threadIdx.x;
  if (c >= nchunks) return;
  float4 tv = reinterpret_cast<const float4*>(t)[c];
  int base = c * 4;
  int4 lv;
  lv.x = (tv.x > 0.0f) ? (base + 0) : -1;
  lv.y = (tv.y > 0.0f) ? (base + 1) : -1;
  lv.z = (tv.z > 0.0f) ? (base + 2) : -1;
  lv.w = (tv.w > 0.0f) ? (base + 3) : -1;
  reinterpret_cast<int4*>(lab)[c] = lv;
  reinterpret_cast<uint4*>(cnt)[c] = make_uint4(0u, 0u, 0u, 0u);
}

// ---- K2: one-pass atomic union-find merge over left/up edges ---------------
__global__ void k_merge(int* __restrict__ lab, int n) {
  int i = blockIdx.x * blockDim.x + threadIdx.x;
  if (i >= n) return;
  if (lab[i] < 0) return;                      // background
  int rem = i & (CCL_HW - 1);
  int row = rem >> 9;                          // / 512
  int col = rem & (CCL_W - 1);
  if (col > 0 && lab[i - 1] >= 0)      unite(lab, i, i - 1);
  if (row > 0 && lab[i - CCL_W] >= 0)  unite(lab, i, i - CCL_W);
}

// ---- K3: flatten to root + accumulate component area -----------------------
__global__ void k_flatten_count(int* __restrict__ lab, unsigned* __restrict__ cnt,
                                int n) {
  int i = blockIdx.x * blockDim.x + threadIdx.x;
  if (i >= n) return;
  if (lab[i] < 0) return;
  int r = find_root(lab, i);
  lab[i] = r;
  atomicAdd(&cnt[r], 1u);
}

// ---- K4: fused BCE + weighting pass, per-block partial sums ----------------
//  s1 = sum_fg bce/(sqrt(area)+1)   s2 = sum_bg bce
//  s3 = sum_fg sqrt(area)           s4 = count_fg
__global__ void k_loss(const float* __restrict__ x, const float* __restrict__ t,
                       const int* __restrict__ lab, const unsigned* __restrict__ cnt,
                       float* __restrict__ part, int nchunks) {
  int tid = blockIdx.x * blockDim.x + threadIdx.x;
  int stride = gridDim.x * blockDim.x;
  float s1 = 0.f, s2 = 0.f, s3 = 0.f, s4 = 0.f;

  for (int c = tid; c < nchunks; c += stride) {
    if (c + stride < nchunks) {                     // speculative prefetch
      __builtin_prefetch(&x[(size_t)(c + stride) * 4], 0, 0);
      __builtin_prefetch(&t[(size_t)(c + stride) * 4], 0, 0);
    }
    float4 xv = reinterpret_cast<const float4*>(x)[c];
    float4 tv = reinterpret_cast<const float4*>(t)[c];
    int4   lv = reinterpret_cast<const int4*>(lab)[c];
    const float* xe = &xv.x;
    const float* te = &tv.x;
    const int*   le = &lv.x;
#pragma unroll
    for (int e = 0; e < 4; ++e) {
      float xx = xe[e], tt = te[e];
      float bce = fmaxf(xx, 0.0f) - xx * tt + log1pf(expf(-fabsf(xx)));
      int li = le[e];
      if (li >= 0) {
        float s = sqrtf((float)cnt[li]);
        s1 += bce / (s + 1.0f);
        s3 += s;
        s4 += 1.0f;
      } else {
        s2 += bce;
      }
    }
  }

  __shared__ float red[4][256];
  red[0][threadIdx.x] = s1;
  red[1][threadIdx.x] = s2;
  red[2][threadIdx.x] = s3;
  red[3][threadIdx.x] = s4;
  __syncthreads();
  for (int s = 128; s > 0; s >>= 1) {
    if (threadIdx.x < (unsigned)s) {
#pragma unroll
      for (int q = 0; q < 4; ++q)
        red[q][threadIdx.x] += red[q][threadIdx.x + s];
    }
    __syncthreads();
  }
  if (threadIdx.x == 0) {
#pragma unroll
    for (int q = 0; q < 4; ++q)
      part[q * NB_PART + blockIdx.x] = red[q][0];
  }
}

// ---- K5: final reduction via V_WMMA_F32_16X16X4_F32 ------------------------
// Single wave32. A (16x4 f32): lane 0-15 hold K=0,1 ; lane 16-31 hold K=2,3.
// Row m -> statistic q = m>>2, sub-row r = m&3. B = all-ones => D row sums.
__global__ void k_final(const float* __restrict__ part, float* __restrict__ out,
                        float invN) {
  int lane = threadIdx.x;            // 0..31, fully active wave
  int m    = lane & 15;
  int kk   = (lane < 16) ? 0 : 2;
  int q    = m >> 2;
  int r    = m & 3;
  const float* plane = part + q * NB_PART;

  v2f bones; bones.x = 1.0f; bones.y = 1.0f;
  v8f c = {};
  const int iters = NB_PART / 16;
  for (int it = 0; it < iters; ++it) {
    int base = it * 16 + r * 4 + kk;
    v2f a;
    a.x = plane[base];
    a.y = plane[base + 1];
    // D[m,n] = sum_k A[m,k] * 1 + C[m,n]  (exact fp32 accumulation)
    c = __builtin_amdgcn_wmma_f32_16x16x4_f32(
        /*neg_a=*/false, a, /*neg_b=*/false, bones,
        /*c_mod=*/(short)0, c, /*reuse_a=*/false, /*reuse_b=*/false);
  }

  // C layout: VGPR v, lanes 0-15 -> row v ; lanes 16-31 -> row v+8.
  __shared__ float rowsum[16];
  int col = lane & 15;
#pragma unroll
  for (int v = 0; v < 8; ++v) {
    int row = (lane < 16) ? v : (v + 8);
    if (col == 0) rowsum[row] = c[v];
  }
  __syncthreads();
  if (lane == 0) {
    float S[4];
#pragma unroll
    for (int qq = 0; qq < 4; ++qq)
      S[qq] = rowsum[4 * qq] + rowsum[4 * qq + 1] +
              rowsum[4 * qq + 2] + rowsum[4 * qq + 3];
    float mean_nz = S[2] / fmaxf(S[3], 1.0f);
    float loss = (S[0] + S[1] / (mean_nz + 1.0f)) * invN;
    out[0] = loss;
  }
}

// ---------------------------------------------------------------------------
extern "C" void kernel_launch(void* const* d_in, const int* in_sizes, int n_in,
                              void* d_out, int out_size, void* d_ws, size_t ws_size,
                              hipStream_t stream) {
  const float* x = (const float*)d_in[0];   // inputs  (logits)
  const float* t = (const float*)d_in[1];   // targets (binary mask)
  int N = in_sizes[0];                      // 32*512*512 = 8388608
  int nchunks = N / 4;

  int*      labels = (int*)d_ws;
  unsigned* counts = (unsigned*)((char*)d_ws + (size_t)N * 4);
  float*    part   = (float*)((char*)d_ws + (size_t)N * 8);

  k_init<<<nchunks / 256, 256, 0, stream>>>(t, labels, counts, nchunks);
  k_merge<<<N / 256, 256, 0, stream>>>(labels, N);
  k_flatten_count<<<N / 256, 256, 0, stream>>>(labels, counts, N);
  k_loss<<<NB_PART, 256, 0, stream>>>(x, t, labels, counts, part, nchunks);
  k_final<<<1, 32, 0, stream>>>(part, (float*)d_out, 1.0f / (float)N);
}